// Gaussian_prev_37632503447674
// MI455X (gfx1250) — compile-verified
//
#include <hip/hip_runtime.h>
#include <hip/hip_bf16.h>

// Problem constants (fixed by the reference)
#define BATCH 8
#define FDIM  64
#define NDIM  2048
#define TOTAL (BATCH * FDIM * NDIM)   // 1,048,576 elements of emb_in

typedef __attribute__((ext_vector_type(2))) float v2f;
typedef __attribute__((ext_vector_type(8))) float v8f;

// ---------------------------------------------------------------------------
// Kernel 1: per-block partial (sum, sumsq) over emb_in. 256 blocks x 256 thr,
// 16 elements/thread. Fixed-order tree reduction => deterministic.
// ---------------------------------------------------------------------------
__global__ __launch_bounds__(256)
void gk_reduce1(const float* __restrict__ emb, float* __restrict__ partials) {
    __shared__ float s0[256];
    __shared__ float s1[256];
    const int tid = threadIdx.x;
    const size_t base = (size_t)blockIdx.x * 256 + tid;
    float s = 0.f, q = 0.f;
#pragma unroll
    for (int k = 0; k < 16; ++k) {
        float v = emb[base + (size_t)k * 65536];
        s += v;
        q += v * v;
    }
    s0[tid] = s;
    s1[tid] = q;
    __syncthreads();
    for (int off = 128; off > 0; off >>= 1) {
        if (tid < off) {
            s0[tid] += s0[tid + off];
            s1[tid] += s1[tid + off];
        }
        __syncthreads();
    }
    if (tid == 0) {
        partials[2 * blockIdx.x]     = s0[0];
        partials[2 * blockIdx.x + 1] = s1[0];
    }
}

// ---------------------------------------------------------------------------
// Kernel 2: fold 256 partials in double, compute
//   var   = (sumsq - sum^2/n) / (n-1)      (ddof=1, matches jnp.std default)
//   scale = 2 / (var * FDIM * SIGMA) = 2 / (128 * var)
// The main kernel consumes RAW emb (no per-element scaling); the augmented
// WMMA produces D = g - (sm+sn)/2 in emb units, and out = exp(D * scale).
// ---------------------------------------------------------------------------
__global__ __launch_bounds__(256)
void gk_reduce2(const float* __restrict__ partials, float* __restrict__ wsf) {
    __shared__ double s0[256];
    __shared__ double s1[256];
    const int tid = threadIdx.x;
    s0[tid] = (double)partials[2 * tid];
    s1[tid] = (double)partials[2 * tid + 1];
    __syncthreads();
    for (int off = 128; off > 0; off >>= 1) {
        if (tid < off) {
            s0[tid] += s0[tid + off];
            s1[tid] += s1[tid + off];
        }
        __syncthreads();
    }
    if (tid == 0) {
        const double n = (double)TOTAL;
        double var = (s1[0] - s0[0] * s0[0] / n) / (n - 1.0);
        wsf[0] = (float)(2.0 / (var * 128.0));   // 128 = FDIM * SIGMA
    }
}

// ---------------------------------------------------------------------------
// Kernel 3: batched C^T C + Gaussian, 32x32 output region per wave (2x2 tiles).
//
// f32 A fragment (16x4):  lane L, VGPR j -> M = L&15, K = (L>>4)*2 + j
// f32 B fragment (4x16):  lane L, VGPR j -> N = L&15, K = (L>>4)*2 + j
// f32 C/D (16x16):        lane L, VGPR v -> N = L&15, M = v + 8*(L>>4)
//
// K is augmented from 64 to 68: the 17th k-block carries
//   A[m][64] = -sm/2, A[m][65] = -1/2, B[64][n] = 1, B[65][n] = sn
// (K=66,67 zero), so D = gram - (sm+sn)/2 and the epilogue is one mul + exp.
// sr/sc are lane-aligned with the fragment layouts (lane = M resp. N), so no
// cross-lane shuffles are needed beyond one shfl_xor(16) completing each norm.
// ---------------------------------------------------------------------------
__global__ __launch_bounds__(256)
void gk_main(const float* __restrict__ emb, const float* __restrict__ wsf,
             float* __restrict__ out) {
    const int lane = threadIdx.x & 31;
    const int wave = threadIdx.x >> 5;

    const int region = blockIdx.x * 8 + wave;      // 32768 regions of 32x32
    const int b      = region >> 12;               // 4096 regions per batch
    const int rem    = region & 4095;
    const int i0     = (rem >> 6) << 5;            // row base (32-aligned)
    const int j0     = (rem & 63) << 5;            // col base (32-aligned)

    const float* E = emb + (size_t)b * FDIM * NDIM;
    const int r  = lane & 15;
    const int kk = (lane >> 4) << 1;               // 0 or 2
    const bool lo = (kk == 0);

    const float* pa = E + (size_t)kk * NDIM + (i0 + r);
    const float* pb = E + (size_t)kk * NDIM + (j0 + r);

    v2f a0[16], a1[16], b0[16], b1[16];
#pragma unroll
    for (int kb = 0; kb < 16; ++kb) {
        const size_t o0 = (size_t)(4 * kb) * NDIM;
        const size_t o1 = (size_t)(4 * kb + 1) * NDIM;
        a0[kb].x = pa[o0];      a0[kb].y = pa[o1];
        a1[kb].x = pa[o0 + 16]; a1[kb].y = pa[o1 + 16];
        b0[kb].x = pb[o0];      b0[kb].y = pb[o1];
        b1[kb].x = pb[o0 + 16]; b1[kb].y = pb[o1 + 16];
    }

    // Squared norms via packed fma chains (each lane: half the features).
    v2f n0 = a0[0] * a0[0], n1 = a1[0] * a1[0];
    v2f m0 = b0[0] * b0[0], m1 = b1[0] * b1[0];
#pragma unroll
    for (int kb = 1; kb < 16; ++kb) {
        n0 += a0[kb] * a0[kb];
        n1 += a1[kb] * a1[kb];
        m0 += b0[kb] * b0[kb];
        m1 += b1[kb] * b1[kb];
    }
    float p0 = n0.x + n0.y, p1 = n1.x + n1.y;
    float q0 = m0.x + m0.y, q1 = m1.x + m1.y;
    const float sr0 = p0 + __shfl_xor(p0, 16, 32); // |row i0+r|^2
    const float sr1 = p1 + __shfl_xor(p1, 16, 32); // |row i0+16+r|^2
    const float sc0 = q0 + __shfl_xor(q0, 16, 32); // |col j0+(lane&15)|^2
    const float sc1 = q1 + __shfl_xor(q1, 16, 32); // |col j0+16+(lane&15)|^2

    // Augmented 17th k-block fragments (K=64..67).
    v2f ax0, ax1, bx0, bx1;
    ax0.x = lo ? (-0.5f * sr0) : 0.f;  ax0.y = lo ? -0.5f : 0.f;
    ax1.x = lo ? (-0.5f * sr1) : 0.f;  ax1.y = lo ? -0.5f : 0.f;
    bx0.x = lo ? 1.f : 0.f;            bx0.y = lo ? sc0 : 0.f;
    bx1.x = lo ? 1.f : 0.f;            bx1.y = lo ? sc1 : 0.f;

    // Four independent accumulator chains -> 4-way WMMA ILP.
    v8f c00 = {}, c01 = {}, c10 = {}, c11 = {};
#pragma unroll
    for (int kb = 0; kb < 16; ++kb) {
        c00 = __builtin_amdgcn_wmma_f32_16x16x4_f32(false, a0[kb], false, b0[kb], (short)0, c00, false, false);
        c01 = __builtin_amdgcn_wmma_f32_16x16x4_f32(false, a0[kb], false, b1[kb], (short)0, c01, false, false);
        c10 = __builtin_amdgcn_wmma_f32_16x16x4_f32(false, a1[kb], false, b0[kb], (short)0, c10, false, false);
        c11 = __builtin_amdgcn_wmma_f32_16x16x4_f32(false, a1[kb], false, b1[kb], (short)0, c11, false, false);
    }
    c00 = __builtin_amdgcn_wmma_f32_16x16x4_f32(false, ax0, false, bx0, (short)0, c00, false, false);
    c01 = __builtin_amdgcn_wmma_f32_16x16x4_f32(false, ax0, false, bx1, (short)0, c01, false, false);
    c10 = __builtin_amdgcn_wmma_f32_16x16x4_f32(false, ax1, false, bx0, (short)0, c10, false, false);
    c11 = __builtin_amdgcn_wmma_f32_16x16x4_f32(false, ax1, false, bx1, (short)0, c11, false, false);

    const float k2 = wsf[0];                       // 2/(var*F*sigma)
    const int hi = (lane & 16) >> 1;               // 0 or 8
    const int n  = lane & 15;
    float* obase = out + (size_t)b * NDIM * NDIM + (size_t)i0 * NDIM + j0;

#pragma unroll
    for (int v = 0; v < 8; ++v) {
        const int m = v + hi;
        obase[(size_t)m * NDIM + n]                   = __expf(c00[v] * k2);
        obase[(size_t)m * NDIM + 16 + n]              = __expf(c01[v] * k2);
        obase[(size_t)(m + 16) * NDIM + n]            = __expf(c10[v] * k2);
        obase[(size_t)(m + 16) * NDIM + 16 + n]       = __expf(c11[v] * k2);
    }
}

// ---------------------------------------------------------------------------
// Launch: d_in[0]=adj_in (unused by reference), d_in[1]=emb_in [8,64,2048] f32.
// d_out: [8,2048,2048] f32. d_ws: wsf[0]=scale, wsf[4..] = 256 float2 partials
// (2064 bytes used).
// ---------------------------------------------------------------------------
extern "C" void kernel_launch(void* const* d_in, const int* in_sizes, int n_in,
                              void* d_out, int out_size, void* d_ws, size_t ws_size,
                              hipStream_t stream) {
    (void)in_sizes; (void)n_in; (void)out_size; (void)ws_size;
    const float* emb = (const float*)d_in[1];
    float* out = (float*)d_out;
    float* wsf = (float*)d_ws;
    float* partials = wsf + 4;                     // 16-byte aligned region

    gk_reduce1<<<256, 256, 0, stream>>>(emb, partials);
    gk_reduce2<<<1, 256, 0, stream>>>(partials, wsf);
    gk_main<<<4096, 256, 0, stream>>>(emb, wsf, out);
}